// RGCN_2791728742679
// MI455X (gfx1250) — compile-verified
//
#include <hip/hip_runtime.h>

// ---- problem constants (match reference) ----
#define N_NODES 100000
#define R_REL   4
#define E_EDGES 400000
#define F_IN    128
#define F_HID   256
#define F_OUT   128

#define KCHUNK  128   // k-values staged in LDS per iteration

typedef float v2f __attribute__((ext_vector_type(2)));
typedef float v8f __attribute__((ext_vector_type(8)));

// ---------------------------------------------------------------------------
// Degree accumulation: deg_out[r][src], deg_in[r][dst] += 1
// ---------------------------------------------------------------------------
__global__ void rgcn_degree_kernel(const int* __restrict__ src,
                                   const int* __restrict__ dst,
                                   float* __restrict__ deg_out,
                                   float* __restrict__ deg_in) {
  int tid = blockIdx.x * blockDim.x + threadIdx.x;
  const int total = R_REL * E_EDGES;
  if (tid >= total) return;
  int r = tid / E_EDGES;
  atomicAdd(&deg_out[(long)r * N_NODES + src[tid]], 1.0f);
  atomicAdd(&deg_in [(long)r * N_NODES + dst[tid]], 1.0f);
}

// in-place: p[i] = rsqrt(max(p[i], 1))
__global__ void rgcn_rsqrt_kernel(float* __restrict__ p, int n) {
  int tid = blockIdx.x * blockDim.x + threadIdx.x;
  if (tid >= n) return;
  float d = p[tid];
  d = d < 1.0f ? 1.0f : d;
  p[tid] = rsqrtf(d);
}

// ---------------------------------------------------------------------------
// Edge scatter: agg[dst[e]] += x[src[e]] * rs_out[src[e]]  (float4 chunks)
// f4shift: log2(F/4)  (F=128 -> 5, F=256 -> 6). agg stays L2-resident.
// ---------------------------------------------------------------------------
__global__ void rgcn_scatter_kernel(const float* __restrict__ x,
                                    const int* __restrict__ s,
                                    const int* __restrict__ d,
                                    const float* __restrict__ rs_out,
                                    float* __restrict__ agg,
                                    int f4shift) {
  const int  F4    = 1 << f4shift;
  const long total = (long)E_EDGES << f4shift;
  const long step  = (long)gridDim.x * blockDim.x;
  for (long tid = (long)blockIdx.x * blockDim.x + threadIdx.x; tid < total; tid += step) {
    int e  = (int)(tid >> f4shift);
    int c  = (int)(tid & (F4 - 1));
    int sv = s[e];
    int dv = d[e];
    float sc = rs_out[sv];
    float4 v = reinterpret_cast<const float4*>(x)[(long)sv * F4 + c];
    float* base = agg + (((long)dv * F4 + c) << 2);
    atomicAdd(base + 0, v.x * sc);
    atomicAdd(base + 1, v.y * sc);
    atomicAdd(base + 2, v.z * sc);
    atomicAdd(base + 3, v.w * sc);
  }
}

// ---------------------------------------------------------------------------
// WMMA fp32 GEMM with accumulate-in-place and LDS-staged B tile:
//   C[M,Nout] = C + (diag(row_scale) * A[M,K]) @ W[K,Nout] + bias
// Exact fp32 via V_WMMA_F32_16X16X4_F32. One wave -> 16x64 output tile
// (4 accumulators share each A fragment). Block = 4 waves = 64 rows; all 4
// waves share one 64-column B tile staged in LDS (coalesced b128 global
// loads, k-pair-packed layout so a B fragment is one conflict-free b64 read).
// Grid: x = ceil(M/64), y = Nout/64.  K % KCHUNK == 0, Nout % 64 == 0.
// ---------------------------------------------------------------------------
__global__ __launch_bounds__(128)
void rgcn_wmma_gemm_acc(const float* __restrict__ A,
                        const float* __restrict__ row_scale,   // may be null
                        const float* __restrict__ W,
                        const float* __restrict__ bias,
                        float* __restrict__ C,
                        int M, int K, int Nout) {
  // Bs[(k/2)*128 + col*2 + (k&1)] for k in [0,KCHUNK), col in [0,64)
  __shared__ float Bs[(KCHUNK / 2) * 128];               // 32 KB

  const int tid  = threadIdx.x;
  const int lane = tid & 31;
  const int wave = tid >> 5;
  const int l16  = lane & 15;
  const int half = lane >> 4;                   // 0 or 1

  const int m0 = (blockIdx.x * 4 + wave) * 16;
  const int n0 = blockIdx.y * 64;

  // A fragment row (ISA 16x4 f32 layout: M = lane%16, K pair selected by lane/16)
  const int   arow   = m0 + l16;
  const int   arow_c = arow < M ? arow : (M - 1);
  const float amask  = arow < M ? 1.0f : 0.0f;
  const float ascale = amask * (row_scale ? row_scale[arow_c] : 1.0f);
  const float* Arow  = A + (long)arow_c * K;

  // C/D layout: VGPR v -> M = m0 + v + (lane>=16 ? 8 : 0), N = n0 + 16j + lane%16
  const int crow = m0 + half * 8;

  v8f acc[4];
  #pragma unroll
  for (int j = 0; j < 4; ++j) {
    const int col = n0 + j * 16 + l16;
    #pragma unroll
    for (int v = 0; v < 8; ++v) {
      int m  = crow + v;
      int mc = m < M ? m : (M - 1);
      acc[j][v] = C[(long)mc * Nout + col];     // OOB lanes read clamped, never stored
    }
  }

  for (int kc = 0; kc < K; kc += KCHUNK) {
    __syncthreads();                            // protect Bs from previous chunk readers
    // Stage W[kc..kc+KCHUNK-1][n0..n0+63]: KCHUNK*16 float4 loads over 128 threads
    #pragma unroll
    for (int i = tid; i < KCHUNK * 16; i += 128) {
      int k  = i >> 4;                          // k row within chunk
      int c4 = (i & 15) << 2;                   // column group of 4
      float4 w = *reinterpret_cast<const float4*>(&W[(long)(kc + k) * Nout + n0 + c4]);
      float* dp = &Bs[(k >> 1) * 128 + c4 * 2 + (k & 1)];
      dp[0] = w.x; dp[2] = w.y; dp[4] = w.z; dp[6] = w.w;
    }
    __syncthreads();

    for (int k0 = 0; k0 < KCHUNK; k0 += 4) {
      const int kkl = k0 + half * 2;            // local k of this lane's A/B pair (even)
      float2 a2 = *reinterpret_cast<const float2*>(Arow + kc + kkl);
      v2f a;
      a.x = a2.x * ascale;
      a.y = a2.y * ascale;
      const float* brow = &Bs[(kkl >> 1) * 128];
      #pragma unroll
      for (int j = 0; j < 4; ++j) {
        v2f b = *reinterpret_cast<const v2f*>(brow + (j * 16 + l16) * 2);
        acc[j] = __builtin_amdgcn_wmma_f32_16x16x4_f32(
            /*neg_a=*/false, a, /*neg_b=*/false, b,
            /*c_mod=*/(short)0, acc[j], /*reuse_a=*/false, /*reuse_b=*/false);
      }
    }
  }

  #pragma unroll
  for (int j = 0; j < 4; ++j) {
    const int col = n0 + j * 16 + l16;
    const float bv = bias ? bias[col] : 0.0f;
    #pragma unroll
    for (int v = 0; v < 8; ++v) {
      int m = crow + v;
      if (m < M) C[(long)m * Nout + col] = acc[j][v] + bv;
    }
  }
}

__global__ void rgcn_relu_kernel(float* __restrict__ p, long n) {
  long tid = (long)blockIdx.x * blockDim.x + threadIdx.x;
  if (tid >= n) return;
  float v = p[tid];
  p[tid] = v > 0.0f ? v : 0.0f;
}

// ---------------------------------------------------------------------------
// Host-side orchestration (all on `stream`, graph-capture safe)
// ---------------------------------------------------------------------------
extern "C" void kernel_launch(void* const* d_in, const int* in_sizes, int n_in,
                              void* d_out, int out_size, void* d_ws, size_t ws_size,
                              hipStream_t stream) {
  (void)in_sizes; (void)n_in; (void)out_size; (void)ws_size;

  const float* x   = (const float*)d_in[0];  // [N, IN]
  const int*   src = (const int*)  d_in[1];  // [R, E]
  const int*   dst = (const int*)  d_in[2];  // [R, E]
  const float* W1  = (const float*)d_in[3];  // [R, IN, HID]
  const float* b1  = (const float*)d_in[4];  // [R, HID]
  const float* W2  = (const float*)d_in[5];  // [R, HID, HID]
  const float* b2  = (const float*)d_in[6];  // [R, HID]
  const float* Wl  = (const float*)d_in[7];  // [HID, OUT]
  const float* bl  = (const float*)d_in[8];  // [OUT]

  float* out    = (float*)d_out;
  float* hidden = out;                         // [N, HID]
  float* output = out + (long)N_NODES * F_HID; // [N, OUT]

  // workspace layout (floats)
  float* ws     = (float*)d_ws;
  float* rs_out = ws;                                   // [R, N] deg_out -> rsqrt
  float* rs_in  = ws + (long)R_REL * N_NODES;           // [R, N] deg_in  -> rsqrt
  float* agg    = ws + 2L * R_REL * N_NODES;            // [N, 256] per-relation aggregate
  float* h1     = agg + (long)N_NODES * F_HID;          // [N, HID] layer-1 activations

  const dim3 gemmBlock(128);
  const int  mBlocks = (N_NODES + 63) / 64;             // 1563

  // --- degrees -> rsqrt scale tables (graph-only, computed once per launch)
  hipMemsetAsync(rs_out, 0, 2L * R_REL * N_NODES * sizeof(float), stream);
  {
    int total = R_REL * E_EDGES;
    rgcn_degree_kernel<<<(total + 255) / 256, 256, 0, stream>>>(src, dst, rs_out, rs_in);
    int n = 2 * R_REL * N_NODES;
    rgcn_rsqrt_kernel<<<(n + 255) / 256, 256, 0, stream>>>(rs_out, n);
  }

  // --- zero accumulation targets
  hipMemsetAsync(h1, 0, (long)N_NODES * F_HID * sizeof(float), stream);
  hipMemsetAsync(d_out, 0, (long)N_NODES * (F_HID + F_OUT) * sizeof(float), stream);

  // --- layer 1: h1 = relu( sum_r norm-aggregate(x) @ W1_r + b1_r )
  for (int r = 0; r < R_REL; ++r) {
    hipMemsetAsync(agg, 0, (long)N_NODES * F_IN * sizeof(float), stream);
    rgcn_scatter_kernel<<<4096, 256, 0, stream>>>(
        x, src + (long)r * E_EDGES, dst + (long)r * E_EDGES,
        rs_out + (long)r * N_NODES, agg, 5 /* F_IN/4 = 32 */);
    rgcn_wmma_gemm_acc<<<dim3(mBlocks, F_HID / 64), gemmBlock, 0, stream>>>(
        agg, rs_in + (long)r * N_NODES,
        W1 + (long)r * F_IN * F_HID, b1 + (long)r * F_HID,
        h1, N_NODES, F_IN, F_HID);
  }
  {
    long n = (long)N_NODES * F_HID;
    rgcn_relu_kernel<<<(int)((n + 255) / 256), 256, 0, stream>>>(h1, n);
  }

  // --- layer 2: hidden = relu( sum_r norm-aggregate(h1) @ W2_r + b2_r )
  for (int r = 0; r < R_REL; ++r) {
    hipMemsetAsync(agg, 0, (long)N_NODES * F_HID * sizeof(float), stream);
    rgcn_scatter_kernel<<<4096, 256, 0, stream>>>(
        h1, src + (long)r * E_EDGES, dst + (long)r * E_EDGES,
        rs_out + (long)r * N_NODES, agg, 6 /* F_HID/4 = 64 */);
    rgcn_wmma_gemm_acc<<<dim3(mBlocks, F_HID / 64), gemmBlock, 0, stream>>>(
        agg, rs_in + (long)r * N_NODES,
        W2 + (long)r * F_HID * F_HID, b2 + (long)r * F_HID,
        hidden, N_NODES, F_HID, F_HID);
  }
  {
    long n = (long)N_NODES * F_HID;
    rgcn_relu_kernel<<<(int)((n + 255) / 256), 256, 0, stream>>>(hidden, n);
  }

  // --- final linear: output = hidden @ Wl + bl  (output region pre-zeroed)
  rgcn_wmma_gemm_acc<<<dim3(mBlocks, F_OUT / 64), gemmBlock, 0, stream>>>(
      hidden, nullptr, Wl, bl, output, N_NODES, F_HID, F_OUT);
}